// DR_35708358099476
// MI455X (gfx1250) — compile-verified
//
#include <hip/hip_runtime.h>

// Graph recsys: 2-layer LightGCN-style SPMM + intent softmax transform.
// SPMM is L2-resident atomic scatter (38MB table << 192MB L2); intent GEMMs
// use V_WMMA_F32_16X16X4_F32 (full fp32 precision, 16-node tile per wave).

typedef __attribute__((ext_vector_type(2))) float v2f;
typedef __attribute__((ext_vector_type(8))) float v8f;

#define EMB  64
#define NINT 128

__device__ __forceinline__ v8f wmma4(v2f a, v2f b, v8f c) {
  // D = A(16x4 f32) * B(4x16 f32) + C(16x16 f32)
  return __builtin_amdgcn_wmma_f32_16x16x4_f32(
      /*neg_a*/false, a, /*neg_b*/false, b,
      /*c_mod*/(short)0, c, /*reuse_a*/false, /*reuse_b*/false);
}

// ---------------- init: embs = concat(user,item); acc = embs ----------------
__global__ void init_kernel(const float* __restrict__ user,
                            const float* __restrict__ item,
                            float* __restrict__ embs, float* __restrict__ acc,
                            long long userElems, long long total) {
  long long i = (long long)blockIdx.x * blockDim.x + threadIdx.x;
  if (i >= total) return;
  float v = (i < userElems) ? user[i] : item[i - userElems];
  embs[i] = v;
  acc[i]  = v;
}

// ---------------- degree (segment_sum of ones over h_idx) -------------------
__global__ void deg_kernel(const int* __restrict__ h, float* __restrict__ deg,
                           int nEdges) {
  int e = blockIdx.x * blockDim.x + threadIdx.x;
  if (e >= nEdges) return;
  unsafeAtomicAdd(&deg[h[e]], 1.0f);
}

__global__ void dinv_kernel(float* __restrict__ d, int nNodes) {
  int i = blockIdx.x * blockDim.x + threadIdx.x;
  if (i >= nNodes) return;
  float dg = d[i];
  d[i] = (dg > 0.0f) ? rsqrtf(dg) : 0.0f;
}

// ---------------- SPMM: y[h] += dinv[h]*dinv[t] * x[t]  ---------------------
// 16 lanes per edge, one float4 per lane. Table + atomics live in L2.
__global__ void spmm_kernel(const int* __restrict__ h, const int* __restrict__ t,
                            const float* __restrict__ dinv,
                            const float* __restrict__ x, float* __restrict__ y,
                            int nEdges) {
  long long gid = (long long)blockIdx.x * blockDim.x + threadIdx.x;
  int e = (int)(gid >> 4);
  if (e >= nEdges) return;
  int c = ((int)gid & 15) << 2;
  int hh = h[e], tt = t[e];
  float g = dinv[hh] * dinv[tt];
  float4 v = *(const float4*)(x + (size_t)tt * EMB + c);
  float* yp = y + (size_t)hh * EMB + c;
  unsafeAtomicAdd(yp + 0, v.x * g);
  unsafeAtomicAdd(yp + 1, v.y * g);
  unsafeAtomicAdd(yp + 2, v.z * g);
  unsafeAtomicAdd(yp + 3, v.w * g);
}

// ---------------- intent transform + combine --------------------------------
// Per wave: 16-node tile. logits[16,128] = E[16,64] @ W[64,128] via wmma f32;
// rowwise softmax in-register; P -> LDS; out[16,64] = P @ W^T via wmma f32;
// embs = gnn + out (in place); acc += embs.
__global__ void __launch_bounds__(128) intent_kernel(
    const float* __restrict__ W,     // intents [64][128] row-major
    const float* __restrict__ gnn,   // [nNodes][64]
    float* __restrict__ embs,        // [nNodes][64], updated in place
    float* __restrict__ acc,         // [nNodes][64], running sum (d_out)
    int nNodes) {
  __shared__ float Pbuf[4][16 * NINT];   // 32 KB: softmax probs per wave

  const int wave = threadIdx.x >> 5;
  const int lane = threadIdx.x & 31;
  const int l15  = lane & 15;
  const int hi   = lane >> 4;            // half-wave: 0 or 1
  const long long base = ((long long)blockIdx.x * 4 + wave) * 16;

  // A fragments for GEMM1: lane (m, hi) holds E[m][4k+2*hi .. +1]
  long long rowNode = base + l15;
  if (rowNode >= nNodes) rowNode = nNodes - 1;   // clamp (stores are guarded)
  const float* arow = embs + (size_t)rowNode * EMB;

  v2f afrag[16];
#pragma unroll
  for (int kk = 0; kk < 16; ++kk)
    afrag[kk] = *(const v2f*)(arow + 4 * kk + 2 * hi);

  v8f logit[8];
#pragma unroll
  for (int n = 0; n < 8; ++n) logit[n] = (v8f){0,0,0,0,0,0,0,0};

#pragma unroll
  for (int n = 0; n < 8; ++n) {
#pragma unroll
    for (int kk = 0; kk < 16; ++kk) {
      int k0 = 4 * kk + 2 * hi;
      v2f b;
      b.x = W[(k0    ) * NINT + n * 16 + l15];
      b.y = W[(k0 + 1) * NINT + n * 16 + l15];
      logit[n] = wmma4(afrag[kk], b, logit[n]);
    }
  }

  // Softmax over the 128 intents for each row held by this half-wave.
  // C layout: vgpr r / lanes[0..15] = row r, lanes[16..31] = row r+8; the 16
  // columns of a row live in one 16-lane half, so xor 1/2/4/8 reduces per-row.
#pragma unroll
  for (int r = 0; r < 8; ++r) {
    float mx = -3.402823466e+38f;
#pragma unroll
    for (int n = 0; n < 8; ++n) mx = fmaxf(mx, logit[n][r]);
#pragma unroll
    for (int m = 1; m <= 8; m <<= 1) mx = fmaxf(mx, __shfl_xor(mx, m, 32));
    float s = 0.0f;
#pragma unroll
    for (int n = 0; n < 8; ++n) {
      float e = __expf(logit[n][r] - mx);
      logit[n][r] = e;
      s += e;
    }
#pragma unroll
    for (int m = 1; m <= 8; m <<= 1) s += __shfl_xor(s, m, 32);
    float inv = 1.0f / s;
#pragma unroll
    for (int n = 0; n < 8; ++n) logit[n][r] *= inv;
  }

  // Stage P[16][128] to LDS to reshape C-layout -> A-layout for GEMM2.
  float* P = &Pbuf[wave][0];
#pragma unroll
  for (int n = 0; n < 8; ++n)
#pragma unroll
    for (int r = 0; r < 8; ++r)
      P[(r + 8 * hi) * NINT + n * 16 + l15] = logit[n][r];
  __syncthreads();

  // GEMM2: out[16][64] = P[16][128] @ W^T.  B(k=j, n=d) = W[d*128 + j], so the
  // two K rows of a fragment are CONTIGUOUS in memory -> v2f loads.
  v2f a2[32];
#pragma unroll
  for (int kk = 0; kk < 32; ++kk)
    a2[kk] = *(const v2f*)&P[l15 * NINT + 4 * kk + 2 * hi];

  v8f o[4];
#pragma unroll
  for (int n2 = 0; n2 < 4; ++n2) o[n2] = (v8f){0,0,0,0,0,0,0,0};

#pragma unroll
  for (int n2 = 0; n2 < 4; ++n2) {
    int d = n2 * 16 + l15;
#pragma unroll
    for (int kk = 0; kk < 32; ++kk) {
      int k0 = 4 * kk + 2 * hi;
      v2f b = *(const v2f*)(W + (size_t)d * NINT + k0);
      o[n2] = wmma4(a2[kk], b, o[n2]);
    }
  }

  // embs = gnn + intent (in place; each tile only touches its own 16 rows),
  // acc += embs.
#pragma unroll
  for (int n2 = 0; n2 < 4; ++n2) {
#pragma unroll
    for (int r = 0; r < 8; ++r) {
      long long node = base + r + 8 * hi;
      if (node < nNodes) {
        int d = n2 * 16 + l15;
        size_t idx = (size_t)node * EMB + d;
        float v = o[n2][r] + gnn[idx];
        embs[idx] = v;
        acc[idx] += v;
      }
    }
  }
}

// ---------------- finalize: out = acc / 3 -----------------------------------
__global__ void finalize_kernel(float* __restrict__ acc, long long total) {
  long long i = (long long)blockIdx.x * blockDim.x + threadIdx.x;
  if (i >= total) return;
  acc[i] *= (1.0f / 3.0f);
}

extern "C" void kernel_launch(void* const* d_in, const int* in_sizes, int n_in,
                              void* d_out, int out_size, void* d_ws, size_t ws_size,
                              hipStream_t stream) {
  const float* user = (const float*)d_in[0];
  const float* item = (const float*)d_in[1];
  const float* W    = (const float*)d_in[2];
  const int*   h    = (const int*)d_in[3];
  const int*   t    = (const int*)d_in[4];

  const int nUsers = in_sizes[0] / EMB;
  const int nItems = in_sizes[1] / EMB;
  const int nNodes = nUsers + nItems;
  const int nEdges = in_sizes[3];
  const long long total = (long long)nNodes * EMB;

  float* acc  = (float*)d_out;                 // running layer sum lives in out
  float* embs = (float*)d_ws;                  // nNodes*64
  float* gnn  = embs + (size_t)nNodes * EMB;   // nNodes*64
  float* dinv = gnn  + (size_t)nNodes * EMB;   // nNodes

  // Layer 0: embs = concat(user,item); acc = embs
  init_kernel<<<(int)((total + 255) / 256), 256, 0, stream>>>(
      user, item, embs, acc, (long long)nUsers * EMB, total);

  // Degrees -> d^{-1/2}
  hipMemsetAsync(dinv, 0, (size_t)nNodes * sizeof(float), stream);
  deg_kernel<<<(nEdges + 255) / 256, 256, 0, stream>>>(h, dinv, nEdges);
  dinv_kernel<<<(nNodes + 255) / 256, 256, 0, stream>>>(dinv, nNodes);

  const int nTiles       = (nNodes + 15) / 16;
  const int intentBlocks = (nTiles + 3) / 4;

  for (int layer = 0; layer < 2; ++layer) {
    hipMemsetAsync(gnn, 0, (size_t)nNodes * EMB * sizeof(float), stream);
    long long spmmThreads = (long long)nEdges * 16;
    spmm_kernel<<<(int)((spmmThreads + 255) / 256), 256, 0, stream>>>(
        h, t, dinv, embs, gnn, nEdges);
    intent_kernel<<<intentBlocks, 128, 0, stream>>>(W, gnn, embs, acc, nNodes);
  }

  finalize_kernel<<<(int)((total + 255) / 256), 256, 0, stream>>>(acc, total);
}